// MinimalMLPEncoder_39178691674345
// MI455X (gfx1250) — compile-verified
//
#include <hip/hip_runtime.h>

typedef __attribute__((ext_vector_type(2))) float v2f;
typedef __attribute__((ext_vector_type(8))) float v8f;

#define KPAD   192   // 180 combos padded to multiple of 4
#define NCOMBO 180
#define HID_OUT 128
#define LDSK   196   // LDS row pitch (floats): 196 % 64 == 4 -> no bank conflicts

// ---------------------------------------------------------------------------
// 0) zero workspace region (counts + T) -- must run every call (graph replay)
// ---------------------------------------------------------------------------
__global__ void zero_kernel(unsigned int* p, long long n) {
    long long i = (long long)blockIdx.x * blockDim.x + threadIdx.x;
    if (i < n) p[i] = 0u;
}

// ---------------------------------------------------------------------------
// 1) Build the 180x128 MLP output table. One block per combo, 128 threads.
// ---------------------------------------------------------------------------
__global__ void table_kernel(const float* __restrict__ emb,
                             const float* __restrict__ dep,
                             const float* __restrict__ W1, const float* __restrict__ b1,
                             const float* __restrict__ W2, const float* __restrict__ b2,
                             const float* __restrict__ W3, const float* __restrict__ b3,
                             float* __restrict__ T) {
    __shared__ float in16[16];
    __shared__ float h1[32];
    __shared__ float h2[64];
    const int c = blockIdx.x;           // 0..179
    const int node = c / 20;
    const int depth = c % 20;
    const int t = threadIdx.x;          // 0..127

    if (t < 8)        in16[t] = emb[node * 8 + t];
    else if (t < 16)  in16[t] = dep[depth * 8 + (t - 8)];
    __syncthreads();

    if (t < 32) {
        float s = b1[t];
        #pragma unroll
        for (int i = 0; i < 16; ++i) s += in16[i] * W1[i * 32 + t];
        h1[t] = fmaxf(s, 0.0f);
    }
    __syncthreads();

    if (t < 64) {
        float s = b2[t];
        #pragma unroll
        for (int i = 0; i < 32; ++i) s += h1[i] * W2[i * 64 + t];
        h2[t] = fmaxf(s, 0.0f);
    }
    __syncthreads();

    float s = b3[t];
    #pragma unroll
    for (int i = 0; i < 64; ++i) s += h2[i] * W3[i * 128 + t];
    T[c * HID_OUT + t] = s;
}

// ---------------------------------------------------------------------------
// 2) Per-(graph, combo) histogram: 2M u32 atomics into L2-resident 12.6MB.
// ---------------------------------------------------------------------------
__global__ void hist_kernel(const int* __restrict__ x,
                            const int* __restrict__ batch,
                            unsigned int* __restrict__ counts,
                            int n) {
    int i = blockIdx.x * blockDim.x + threadIdx.x;
    if (i >= n) return;
    int node = x[2 * i];
    int d    = x[2 * i + 1];
    d = min(max(d, 0), 19);
    int c = node * 20 + d;
    atomicAdd(&counts[(long long)batch[i] * KPAD + c], 1u);
}

// ---------------------------------------------------------------------------
// 3) In-place u32 -> f32 conversion of counts (exact: counts << 2^24).
// ---------------------------------------------------------------------------
__global__ void cvt_kernel(unsigned int* p, long long n) {
    long long i = (long long)blockIdx.x * blockDim.x + threadIdx.x;
    if (i < n) {
        float v = (float)p[i];
        p[i] = __float_as_uint(v);
    }
}

// ---------------------------------------------------------------------------
// 4) out = diag(1/n_g) * C * T via V_WMMA_F32_16X16X4_F32.
//    Block = 256 threads = 8 wave32; one block per 16-graph M-tile;
//    wave w handles N-tile [16w, 16w+16). Counts tile staged in LDS.
// ---------------------------------------------------------------------------
__global__ void wmma_mean_kernel(const float* __restrict__ counts,
                                 const float* __restrict__ T,
                                 float* __restrict__ out) {
    __shared__ float sA[16 * LDSK];
    __shared__ float sInv[16];

    const int g0  = blockIdx.x * 16;
    const int tid = threadIdx.x;

    // Cooperative stage of the 16 x 192 counts tile into LDS.
    for (int i = tid; i < 16 * KPAD; i += 256) {
        int r = i / KPAD;
        int k = i % KPAD;
        sA[r * LDSK + k] = counts[(long long)(g0 + r) * KPAD + k];
    }
    __syncthreads();

    // Per-graph 1/max(n,1): n is the row-sum of the counts tile.
    if (tid < 16) {
        float s = 0.0f;
        for (int k = 0; k < NCOMBO; ++k) s += sA[tid * LDSK + k];
        sInv[tid] = 1.0f / fmaxf(s, 1.0f);
    }
    __syncthreads();

    const int wave = tid >> 5;          // 0..7  -> N-tile
    const int lane = tid & 31;
    const int n0   = wave * 16;
    const int h    = lane >> 4;         // lane half: 0 or 1
    const int r    = lane & 15;         // A row (M) / B,D column (N)

    v8f acc = {};
    for (int k0 = 0; k0 < KPAD; k0 += 4) {
        // A 16x4 f32: lanes 0-15 K={k0,k0+1}, lanes 16-31 K={k0+2,k0+3}
        v2f a, b;
        a.x = sA[r * LDSK + k0 + 2 * h + 0];
        a.y = sA[r * LDSK + k0 + 2 * h + 1];
        // B 4x16 f32: VGPR v holds row K=k0+v (lanes 0-15) / K=k0+v+2 (lanes 16-31)
        b.x = T[(k0 + 2 * h + 0) * HID_OUT + n0 + r];
        b.y = T[(k0 + 2 * h + 1) * HID_OUT + n0 + r];
        acc = __builtin_amdgcn_wmma_f32_16x16x4_f32(
                  /*neg_a=*/false, a, /*neg_b=*/false, b,
                  /*c_mod=*/(short)0, acc, /*reuse_a=*/false, /*reuse_b=*/false);
    }

    // D layout: VGPR v -> M = v + 8*h, N = lane%16. Apply mean scale and store.
    #pragma unroll
    for (int v = 0; v < 8; ++v) {
        int m = v + 8 * h;
        out[(long long)(g0 + m) * HID_OUT + n0 + r] = acc[v] * sInv[m];
    }
}

// ---------------------------------------------------------------------------
extern "C" void kernel_launch(void* const* d_in, const int* in_sizes, int n_in,
                              void* d_out, int out_size, void* d_ws, size_t ws_size,
                              hipStream_t stream) {
    const int*   x       = (const int*)d_in[0];
    const int*   batch   = (const int*)d_in[1];
    // d_in[2] = num_graphs (device scalar; value derived from out_size instead)
    const float* emb     = (const float*)d_in[3];
    const float* dep     = (const float*)d_in[4];
    const float* W1      = (const float*)d_in[5];
    const float* b1      = (const float*)d_in[6];
    const float* W2      = (const float*)d_in[7];
    const float* b2      = (const float*)d_in[8];
    const float* W3      = (const float*)d_in[9];
    const float* b3      = (const float*)d_in[10];
    float*       out     = (float*)d_out;

    const int n_nodes = in_sizes[0] / 2;
    const int n_graphs = out_size / HID_OUT;          // 16384

    // Workspace layout: [counts: G*192 f32][T: 192*128 f32]
    unsigned int* d_counts_u = (unsigned int*)d_ws;
    float*        d_counts_f = (float*)d_ws;
    float*        d_T        = (float*)d_ws + (long long)n_graphs * KPAD;

    const long long zero_n = (long long)n_graphs * KPAD + (long long)KPAD * HID_OUT;

    // 0) zero counts + T (graph-replay safe)
    {
        long long blocks = (zero_n + 255) / 256;
        zero_kernel<<<(unsigned)blocks, 256, 0, stream>>>((unsigned int*)d_ws, zero_n);
    }
    // 1) MLP table over the 180 distinct inputs
    table_kernel<<<NCOMBO, 128, 0, stream>>>(emb, dep, W1, b1, W2, b2, W3, b3, d_T);
    // 2) histogram
    hist_kernel<<<(n_nodes + 255) / 256, 256, 0, stream>>>(x, batch, d_counts_u, n_nodes);
    // 3) counts u32 -> f32 in place
    {
        long long cn = (long long)n_graphs * KPAD;
        cvt_kernel<<<(unsigned)((cn + 255) / 256), 256, 0, stream>>>(d_counts_u, cn);
    }
    // 4) WMMA GEMM + mean
    wmma_mean_kernel<<<n_graphs / 16, 256, 0, stream>>>(d_counts_f, d_T, out);
}